// GPTQMarlin24Linear_58128087384231
// MI455X (gfx1250) — compile-verified
//
#include <hip/hip_runtime.h>
#include <stdint.h>

// ---------------------------------------------------------------------------
// GPTQ Marlin 2:4 int4 GEMM for gfx1250 (MI455X) — V_SWMMAC_F32_16X16X64_BF16.
//   out[M,N] = A[M,K] * dequant24(B,meta,s)[K,N] + bias[N]
// W is 2:4 sparse along K: feed it as the SWMMAC sparse A-operand (W^T tiles,
// stored compressed at half size) with meta translated directly into the
// hardware 2-bit index nibbles. Activations (fp32 -> bf16) are the dense
// B-operand. D = W^T x A^T = (A x W)^T per 16x16 tile.
// ---------------------------------------------------------------------------

typedef __attribute__((ext_vector_type(16))) __bf16        v16bf;
typedef __attribute__((ext_vector_type(32))) __bf16        v32bf;
typedef __attribute__((ext_vector_type(2)))  __bf16        v2bf;
typedef __attribute__((ext_vector_type(8)))  float         v8f;
typedef __attribute__((ext_vector_type(4)))  unsigned int  u32x4;
typedef __attribute__((ext_vector_type(2)))  unsigned int  u32x2;

#define BLK_M 128
#define BLK_N 64
#define BLK_K 64    // dense K per slab (32 compressed)
#define ASTR  72    // LDS stride for activation rows: 64 bf16 + pad (144B, 16B-aligned)
#define WSTR  40    // LDS stride for compressed weight rows: 32 bf16 + pad (80B)

// pairs {0,1},{0,2},{0,3},{1,2},{1,3},{2,3} -> nibble = pos0 | pos1<<2
// (exactly the SWMMAC 2-bit index pair format, Idx0 < Idx1 guaranteed)
#define PAIR_TABLE 0x00ED9C84u

// native pack: two f32 -> packed bf16x2 (v_cvt_pk_bf16_f32 on gfx1250)
__device__ __forceinline__ unsigned pkbf(float a, float b) {
  v2bf p; p.x = (__bf16)a; p.y = (__bf16)b;
  return __builtin_bit_cast(unsigned, p);
}

union FragA { u32x4 u[2]; v16bf v; };   // sparse operand: compressed W^T 16x32
union FragB { u32x4 u[4]; v32bf v; };   // dense operand: activations 64x16

__global__ __launch_bounds__(256, 2)
void marlin24_swmmac(const float* __restrict__ A,
                     const int*   __restrict__ Bq,    // [K/16, N] packed nibbles
                     const int*   __restrict__ meta,  // [K/4,  N] in [0,6)
                     const float* __restrict__ s,     // [K/GS, N]
                     const float* __restrict__ bias,  // [N]
                     float*       __restrict__ out,   // [M, N]
                     int M, int N, int K, int GS)
{
  __shared__ unsigned short Ash[BLK_M * ASTR];   // activations, bf16 [M][K]
  __shared__ unsigned short Wc [BLK_N * WSTR];   // compressed weights, bf16 [N][K/2]
  __shared__ unsigned short IdxL[BLK_N * 4];     // index nibbles: 4 ushorts per N row

  const int t    = threadIdx.x;     // 256 threads = 8 wave32
  const int lane = t & 31;
  const int w    = t >> 5;
  const int wm   = w & 3;           // 4 waves along M (32 rows each)
  const int wn   = w >> 2;          // 2 waves along N (32 cols each)

  const int n0 = blockIdx.x * BLK_N;
  const int m0 = blockIdx.y * BLK_M;

  v8f acc[2][2];                    // acc[ni][mi] : D[n16][m16] tiles
#pragma unroll
  for (int ni = 0; ni < 2; ++ni)
#pragma unroll
    for (int mi = 0; mi < 2; ++mi)
#pragma unroll
      for (int i = 0; i < 8; ++i) acc[ni][mi][i] = 0.0f;

  // ---- staging indices ----
  const int arow  = t >> 1;          // activation row, 2 threads per row
  const int ahalf = (t & 1) * 32;    // 32 floats each
  const int dn    = t & 63;          // weight column within tile
  const int dkq   = t >> 6;          // K-quarter (16 dense K each), 0..3

  // ---- fragment addressing ----
  const int lmod = lane & 15;
  const int grp  = lane >> 4;        // lane half selects K-halves per ISA layouts
  const int coff = grp * 8;          // compressed A-frag: c {0..7,16..23}/{8..15,24..31}
  const int boff = grp * 16;         // dense B-frag: K {0..15,32..47}/{16..31,48..63}

  for (int k0 = 0; k0 < K; k0 += BLK_K) {
    __syncthreads();   // previous slab fully consumed

    // ---- stage activations: fp32 -> bf16 into LDS ----
    {
      const float4* src = (const float4*)(A + (size_t)(m0 + arow) * K + k0 + ahalf);
      unsigned short* dst = &Ash[arow * ASTR + ahalf];
#pragma unroll
      for (int i = 0; i < 8; ++i) {
        float4 vv = src[i];
        u32x2 p;
        p.x = pkbf(vv.x, vv.y);
        p.y = pkbf(vv.z, vv.w);
        *(u32x2*)(dst + i * 4) = p;
      }
      if (k0 + BLK_K < K)  // CDNA5 global_prefetch_b8 for next slab
        __builtin_prefetch((const char*)src + BLK_K * sizeof(float));
    }

    // ---- stage compressed weights + hardware index nibbles ----
    {
      const int gcol = n0 + dn;
      const unsigned packed = (unsigned)Bq[(size_t)((k0 >> 4) + dkq) * N + gcol];
      const float scale = s[(size_t)(k0 / GS) * N + gcol];
      const int mrow = (k0 >> 2) + dkq * 4;
      unsigned r[4];
      unsigned idxbits = 0u;
#pragma unroll
      for (int j = 0; j < 4; ++j) {          // 4 groups of 4 dense K
        const int mm = meta[(size_t)(mrow + j) * N + gcol];
        idxbits |= ((PAIR_TABLE >> (mm << 2)) & 0xFu) << (4 * j);
        const float v0 = (float)((int)((packed >> (8 * j))     & 0xF) - 8) * scale;
        const float v1 = (float)((int)((packed >> (8 * j + 4)) & 0xF) - 8) * scale;
        r[j] = pkbf(v0, v1);                 // compressed pair (slot0, slot1)
      }
      u32x4 q; q.x = r[0]; q.y = r[1]; q.z = r[2]; q.w = r[3];
      *(u32x4*)(&Wc[dn * WSTR + dkq * 8]) = q;          // 8 compressed bf16
      IdxL[dn * 4 + dkq] = (unsigned short)idxbits;     // 4 index nibbles
    }

    __syncthreads();   // slab staged

    // ---- compute: 2x2 SWMMAC per wave on this 64-K slab ----
    FragA fw[2];
    int   idxv[2];
#pragma unroll
    for (int ni = 0; ni < 2; ++ni) {
      const int nrow = wn * 32 + ni * 16 + lmod;
      const unsigned short* p = &Wc[nrow * WSTR + coff];
      fw[ni].u[0] = *(const u32x4*)(p);
      fw[ni].u[1] = *(const u32x4*)(p + 16);
      idxv[ni] = *(const int*)(&IdxL[nrow * 4 + grp * 2]);  // groups 0-7 / 8-15
    }
    FragB fb[2];
#pragma unroll
    for (int mi = 0; mi < 2; ++mi) {
      const unsigned short* q = &Ash[(wm * 32 + mi * 16 + lmod) * ASTR + boff];
      fb[mi].u[0] = *(const u32x4*)(q);
      fb[mi].u[1] = *(const u32x4*)(q + 8);
      fb[mi].u[2] = *(const u32x4*)(q + 32);
      fb[mi].u[3] = *(const u32x4*)(q + 40);
    }
#pragma unroll
    for (int ni = 0; ni < 2; ++ni)
#pragma unroll
      for (int mi = 0; mi < 2; ++mi)
        acc[ni][mi] = __builtin_amdgcn_swmmac_f32_16x16x64_bf16(
            /*neg_a=*/false, fw[ni].v, /*neg_b=*/false, fb[mi].v,
            acc[ni][mi], idxv[ni], /*reuse_a=*/false, /*reuse_b=*/false);
  }

  // ---- epilogue: D[n][m] tiles -> out[m][n] + bias (8 consecutive N per lane) ----
#pragma unroll
  for (int ni = 0; ni < 2; ++ni)
#pragma unroll
    for (int mi = 0; mi < 2; ++mi) {
      const int row  = m0 + wm * 32 + mi * 16 + lmod;        // output M (D column)
      const int col0 = n0 + wn * 32 + ni * 16 + grp * 8;     // output N (D row base)
      const float4 b0 = *(const float4*)(bias + col0);
      const float4 b1 = *(const float4*)(bias + col0 + 4);
      float* dst = out + (size_t)row * N + col0;
      float4 o0, o1;
      o0.x = acc[ni][mi][0] + b0.x;  o0.y = acc[ni][mi][1] + b0.y;
      o0.z = acc[ni][mi][2] + b0.z;  o0.w = acc[ni][mi][3] + b0.w;
      o1.x = acc[ni][mi][4] + b1.x;  o1.y = acc[ni][mi][5] + b1.y;
      o1.z = acc[ni][mi][6] + b1.z;  o1.w = acc[ni][mi][7] + b1.w;
      *(float4*)(dst)     = o0;
      *(float4*)(dst + 4) = o1;
    }
}

extern "C" void kernel_launch(void* const* d_in, const int* in_sizes, int n_in,
                              void* d_out, int out_size, void* d_ws, size_t ws_size,
                              hipStream_t stream) {
  const float* A    = (const float*)d_in[0];
  const int*   Bq   = (const int*)d_in[1];
  const int*   meta = (const int*)d_in[2];
  const float* s    = (const float*)d_in[3];
  const float* bias = (const float*)d_in[4];
  float*       out  = (float*)d_out;

  const long long N  = in_sizes[4];                       // bias length
  const long long K  = (16LL * in_sizes[1]) / N;          // B is [K/16, N]
  const long long M  = (long long)in_sizes[0] / K;        // A is [M, K]
  const long long GS = (K * N) / in_sizes[3];             // s is [K/GS, N]

  dim3 grid((unsigned)(N / BLK_N), (unsigned)(M / BLK_M));
  marlin24_swmmac<<<grid, 256, 0, stream>>>(A, Bq, meta, s, bias, out,
                                            (int)M, (int)N, (int)K, (int)GS);
}